// GraphSAGEModel_87050397156004
// MI455X (gfx1250) — compile-verified
//
#include <hip/hip_runtime.h>
#include <hip/hip_bf16.h>

#define D        128
#define DOUT     10
#define NGRAPHS  512

typedef __attribute__((ext_vector_type(2))) float v2f;
typedef __attribute__((ext_vector_type(8))) float v8f;

// ---------------------------------------------------------------- zero fill
__global__ void zero_f32(float* __restrict__ p, long n) {
    long i = (long)blockIdx.x * blockDim.x + threadIdx.x;
    long stride = (long)gridDim.x * blockDim.x;
    for (; i < n; i += stride) p[i] = 0.0f;
}

// ------------------------------------------------- edge gather/scatter (mean aggr numerator)
// one wave per edge: 32 lanes x float4 = 128 features
__global__ void edge_scatter(const float* __restrict__ feat,
                             const int* __restrict__ src,
                             const int* __restrict__ dst,
                             float* __restrict__ accum,
                             float* __restrict__ cnt,   // may be nullptr
                             int n_edges) {
    int wave = (int)((blockIdx.x * blockDim.x + threadIdx.x) >> 5);
    int lane = threadIdx.x & 31;
    if (wave >= n_edges) return;
    int s = src[wave];
    int d = dst[wave];
    const float4 v = *(const float4*)(feat + (long)s * D + lane * 4);
    float* a = accum + (long)d * D + lane * 4;
    atomicAdd(a + 0, v.x);
    atomicAdd(a + 1, v.y);
    atomicAdd(a + 2, v.z);
    atomicAdd(a + 3, v.w);
    if (cnt != nullptr && lane == 0) atomicAdd(cnt + d, 1.0f);
}

// ------------------------------------------------- fused SAGEConv GEMM (fp32 WMMA)
// out[n,o] = relu( (accum[n,:]/max(cnt[n],1)) . Wl[o,:] + bias[o] + self[n,:] . Wr[o,:] )
// wave -> 16x16 tile of out; block (256 thr, 8 waves) -> 16 rows x 128 cols.
// Two independent WMMA accumulator chains (neighbor term / self term) interleaved
// in one K-loop for XDL ILP; the 1/cnt mean-scale is applied once to the
// accumulator after the loop instead of per-fragment.
__global__ void __launch_bounds__(256)
sage_gemm_relu(const float* __restrict__ Am,    // neighbor sums [n,128]
               const float* __restrict__ cnt,   // [n]
               const float* __restrict__ As,    // self features [n,128]
               const float* __restrict__ Wl,    // [128,128] (out,in) row-major
               const float* __restrict__ Wr,    // [128,128]
               const float* __restrict__ bias,  // [128]
               float* __restrict__ out,         // [n,128]
               int n) {
    const int lane    = threadIdx.x & 31;
    const int wave    = threadIdx.x >> 5;        // 0..7 -> 16-col slab
    const int rowbase = blockIdx.x * 16;
    const int r       = lane & 15;
    int row = rowbase + r;
    if (row >= n) row = n - 1;                   // clamp (keeps EXEC all-ones)
    const int col = wave * 16 + r;               // 0..127
    const int klo = (lane < 16) ? 0 : 2;         // A/B fragment K-offset per ISA layout

    const float inv = 1.0f / fmaxf(cnt[row], 1.0f);

    const float* amrow = Am + (long)row * D + klo;   // neighbor-sum row
    const float* wlrow = Wl + (long)col * D + klo;   // B[k,col] = Wl[col,k]
    const float* asrow = As + (long)row * D + klo;   // self row
    const float* wrrow = Wr + (long)col * D + klo;   // B[k,col] = Wr[col,k]

    v8f acc_n = {};   // neighbor-term chain
    v8f acc_s = {};   // self-term chain
#pragma unroll
    for (int k = 0; k < D; k += 4) {
        v2f an = *(const v2f*)(amrow + k);
        v2f bn = *(const v2f*)(wlrow + k);
        v2f as = *(const v2f*)(asrow + k);
        v2f bs = *(const v2f*)(wrrow + k);
        acc_n = __builtin_amdgcn_wmma_f32_16x16x4_f32(
            false, an, false, bn, (short)0, acc_n, false, false);
        acc_s = __builtin_amdgcn_wmma_f32_16x16x4_f32(
            false, as, false, bs, (short)0, acc_s, false, false);
    }

    // D layout: element v of lane -> (M = v + (lane>=16 ? 8 : 0), N = lane&15)
    const float bcol  = bias[col];
    const int   mrow0 = rowbase + ((lane < 16) ? 0 : 8);
#pragma unroll
    for (int v = 0; v < 8; ++v) {
        int orow = mrow0 + v;
        if (orow < n) {
            float val = acc_n[v] * inv + acc_s[v] + bcol;   // mean-scale folded here
            out[(long)orow * D + col] = fmaxf(val, 0.0f);
        }
    }
}

// ------------------------------------------------- global mean pool scatter
// one wave per node
__global__ void pool_scatter(const float* __restrict__ h,
                             const int* __restrict__ batch,
                             float* __restrict__ gsum,
                             float* __restrict__ gcnt,
                             int n) {
    int node = (int)((blockIdx.x * blockDim.x + threadIdx.x) >> 5);
    int lane = threadIdx.x & 31;
    if (node >= n) return;
    int g = batch[node];
    const float4 v = *(const float4*)(h + (long)node * D + lane * 4);
    float* a = gsum + (long)g * D + lane * 4;
    atomicAdd(a + 0, v.x);
    atomicAdd(a + 1, v.y);
    atomicAdd(a + 2, v.z);
    atomicAdd(a + 3, v.w);
    if (lane == 0) atomicAdd(gcnt + g, 1.0f);
}

// ------------------------------------------------- head: linear + log_softmax
// one wave per graph
__global__ void head_kernel(const float* __restrict__ gsum,
                            const float* __restrict__ gcnt,
                            const float* __restrict__ Wf,  // [10,128]
                            const float* __restrict__ bf,  // [10]
                            float* __restrict__ outp) {    // [512,10]
    int g = blockIdx.x;
    int lane = threadIdx.x;                      // blockDim == 32
    float inv = 1.0f / fmaxf(gcnt[g], 1.0f);
    float4 m = *(const float4*)(gsum + (long)g * D + lane * 4);
    m.x *= inv; m.y *= inv; m.z *= inv; m.w *= inv;

    float logit[DOUT];
#pragma unroll
    for (int o = 0; o < DOUT; ++o) {
        const float* w = Wf + (long)o * D + lane * 4;
        float p = m.x * w[0] + m.y * w[1] + m.z * w[2] + m.w * w[3];
#pragma unroll
        for (int s = 16; s > 0; s >>= 1) p += __shfl_xor(p, s, 32);
        logit[o] = p + bf[o];
    }
    if (lane == 0) {
        float mx = logit[0];
#pragma unroll
        for (int o = 1; o < DOUT; ++o) mx = fmaxf(mx, logit[o]);
        float se = 0.0f;
#pragma unroll
        for (int o = 0; o < DOUT; ++o) se += __expf(logit[o] - mx);
        float lse = mx + __logf(se);
#pragma unroll
        for (int o = 0; o < DOUT; ++o) outp[(long)g * DOUT + o] = logit[o] - lse;
    }
}

// ----------------------------------------------------------------------------
extern "C" void kernel_launch(void* const* d_in, const int* in_sizes, int n_in,
                              void* d_out, int out_size, void* d_ws, size_t ws_size,
                              hipStream_t stream) {
    const float* x     = (const float*)d_in[0];
    const int*   ei    = (const int*)d_in[1];    // [2, E] flat
    const int*   batch = (const int*)d_in[2];
    const float* W1l   = (const float*)d_in[3];
    const float* b1    = (const float*)d_in[4];
    const float* W1r   = (const float*)d_in[5];
    const float* W2l   = (const float*)d_in[6];
    const float* b2    = (const float*)d_in[7];
    const float* W2r   = (const float*)d_in[8];
    const float* Wf    = (const float*)d_in[9];
    const float* bf    = (const float*)d_in[10];
    float* out = (float*)d_out;

    const int n = in_sizes[0] / D;               // 100000 nodes
    const int e = in_sizes[1] / 2;               // 1600000 edges
    const int* src = ei;
    const int* dst = ei + e;

    // workspace carve-up (256B aligned)
    char*  ws  = (char*)d_ws;
    size_t off = 0;
    auto carve = [&](size_t bytes) -> float* {
        float* p = (float*)(ws + off);
        off += (bytes + 255) & ~(size_t)255;
        return p;
    };
    float* accum = carve((size_t)n * D * sizeof(float));       // 51.2 MB (reused both layers)
    float* h1    = carve((size_t)n * D * sizeof(float));       // 51.2 MB
    float* h2    = carve((size_t)n * D * sizeof(float));       // 51.2 MB
    float* cnt   = carve((size_t)n * sizeof(float));
    float* gsum  = carve((size_t)NGRAPHS * D * sizeof(float));
    float* gcnt  = carve((size_t)NGRAPHS * sizeof(float));
    (void)ws_size; (void)n_in; (void)out_size;

    const long nd = (long)n * D;
    const int  ZB = 1024;                                      // zero-fill blocks
    dim3 blk256(256), blk32(32);
    int edge_blocks = (int)(((long)e * 32 + 255) / 256);
    int node_blocks = (int)(((long)n * 32 + 255) / 256);
    int gemm_blocks = (n + 15) / 16;

    // ---- layer 1
    zero_f32<<<ZB, blk256, 0, stream>>>(accum, nd);
    zero_f32<<<ZB, blk256, 0, stream>>>(cnt, n);
    edge_scatter<<<edge_blocks, blk256, 0, stream>>>(x, src, dst, accum, cnt, e);
    sage_gemm_relu<<<gemm_blocks, blk256, 0, stream>>>(accum, cnt, x, W1l, W1r, b1, h1, n);

    // ---- layer 2 (counts are identical; reuse cnt)
    zero_f32<<<ZB, blk256, 0, stream>>>(accum, nd);
    edge_scatter<<<edge_blocks, blk256, 0, stream>>>(h1, src, dst, accum, nullptr, e);
    sage_gemm_relu<<<gemm_blocks, blk256, 0, stream>>>(accum, cnt, h1, W2l, W2r, b2, h2, n);

    // ---- global mean pool + head
    zero_f32<<<ZB, blk256, 0, stream>>>(gsum, (long)NGRAPHS * D);
    zero_f32<<<ZB, blk256, 0, stream>>>(gcnt, NGRAPHS);
    pool_scatter<<<node_blocks, blk256, 0, stream>>>(h2, batch, gsum, gcnt, n);
    head_kernel<<<NGRAPHS, blk32, 0, stream>>>(gsum, gcnt, Wf, bf, out);
}